// AlgebraicTriangulation_30468497997957
// MI455X (gfx1250) — compile-verified
//
#include <hip/hip_runtime.h>
#include <hip/hip_bf16.h>
#include <math.h>

// Problem constants from the reference
#define NB 64
#define NC 4
#define NJ 17
#define HM_H 96
#define HM_W 96
#define HM_N (HM_H * HM_W)     // 9216
#define T1 256
#define PER_T (HM_N / T1)      // 36
#define INV_TEMP 100.0f

typedef float v2f_t __attribute__((ext_vector_type(2)));
typedef float v8f_t __attribute__((ext_vector_type(8)));

// ---------------------------------------------------------------------------
// Kernel 1: softargmax over each (bc, j) heatmap. One block per map.
// Bandwidth-bound: single coalesced read pass cached in registers (36 VGPRs),
// max-reduce, then exp/weighted-sum reduce via LDS trees.
// ---------------------------------------------------------------------------
__global__ void __launch_bounds__(T1) softargmax_kernel(
    const float* __restrict__ hm, float* __restrict__ kp,
    const int* __restrict__ Himg, const int* __restrict__ Wimg)
{
  const int map = blockIdx.x;                 // bc * NJ + j
  const float* __restrict__ p = hm + (size_t)map * HM_N;
  const int tid = threadIdx.x;

  float v[PER_T];
  float mx = -3.402823466e38f;
#pragma unroll
  for (int i = 0; i < PER_T; ++i) {
    v[i] = p[tid + i * T1];
    mx = fmaxf(mx, v[i]);
  }

  __shared__ float rmax[T1];
  rmax[tid] = mx;
  __syncthreads();
  for (int s = T1 / 2; s > 0; s >>= 1) {
    if (tid < s) rmax[tid] = fmaxf(rmax[tid], rmax[tid + s]);
    __syncthreads();
  }
  mx = rmax[0];
  __syncthreads();

  float ssum = 0.0f, sx = 0.0f, sy = 0.0f;
#pragma unroll
  for (int i = 0; i < PER_T; ++i) {
    const int idx = tid + i * T1;
    const float e = __expf((v[i] - mx) * INV_TEMP);
    ssum += e;
    sx += e * (float)(idx % HM_W);   // x coordinate (width)
    sy += e * (float)(idx / HM_W);   // y coordinate (height)
  }

  __shared__ float rs[T1], rx[T1], ry[T1];
  rs[tid] = ssum; rx[tid] = sx; ry[tid] = sy;
  __syncthreads();
  for (int s = T1 / 2; s > 0; s >>= 1) {
    if (tid < s) {
      rs[tid] += rs[tid + s];
      rx[tid] += rx[tid + s];
      ry[tid] += ry[tid + s];
    }
    __syncthreads();
  }

  if (tid == 0) {
    const float inv = 1.0f / rs[0];
    // reference: kp = [kx, ky] * [H_img/Hh, W_img/Wh]
    kp[map * 2 + 0] = rx[0] * inv * ((float)Himg[0] / (float)HM_H);
    kp[map * 2 + 1] = ry[0] * inv * ((float)Wimg[0] / (float)HM_W);
  }
}

// ---------------------------------------------------------------------------
// Kernel 2: DLT. One wave (32 lanes) handles 4 (b,j) problems.
// Gram matrices M_p = A_p^T A_p (4x4, from 8x4 A_p) are computed with
// V_WMMA_F32_16X16X4_F32: problems packed on the diagonal 4x4 blocks of the
// 16x16 tile; K=8 split into two accumulated K=4 WMMAs. Because the product
// is X^T X and the A/B operand lane layouts are transposes of each other,
// a single per-lane fragment serves as both WMMA operands.
// Then 4 lanes run a cyclic Jacobi eigensolver on their 4x4 Gram matrix and
// emit eigvec(min lambda)[:3] / eigvec[3].
// ---------------------------------------------------------------------------
__global__ void __launch_bounds__(32) dlt_wmma_kernel(
    const float* __restrict__ P,      // (B, C, 3, 4)
    const float* __restrict__ conf,   // (B, C, J)
    const float* __restrict__ kp,     // (bc, j, 2) == (B, C, J, 2)
    float* __restrict__ out)          // (B, J, 3)
{
  __shared__ float As[4][8][4];   // [problem][row 2c+i][col]
  __shared__ float Gs[4][4][4];   // [problem][r][c] Gram matrices

  const int lane = threadIdx.x;
  const int base = blockIdx.x * 4;

  // Build the four 8x4 A matrices cooperatively: 128 entries, 4 per lane.
  // A[2c+i][col] = (P[b,c,2,col]*kp[b,c,j,i] - P[b,c,i,col]) * conf[b,c,j]
#pragma unroll
  for (int t = 0; t < 4; ++t) {
    const int e = lane + t * 32;
    const int pidx = e >> 5, rem = e & 31;
    const int row = rem >> 2, col = rem & 3;
    const int c = row >> 1, i = row & 1;
    const int g = base + pidx;
    const int b = g / NJ, j = g % NJ;
    const float* Pm = P + (size_t)((b * NC + c) * 3) * 4;
    const int kidx = (b * NC + c) * NJ + j;
    const float k2 = kp[kidx * 2 + i];
    const float w  = conf[kidx];
    As[pidx][row][col] = (Pm[2 * 4 + col] * k2 - Pm[i * 4 + col]) * w;
  }
  __syncthreads();

  // WMMA fragment load. A operand (16x4, MxK): lane L -> M = L&15,
  // VGPR0 holds K = 2*(L>>4), VGPR1 holds K+1. B operand (4x16, KxN) has the
  // transposed-identical per-lane layout, and our B matrix is A^T in WMMA
  // coordinates, so the same fragment is used for both operands.
  const int m  = lane & 15;
  const int k0 = (lane >> 4) << 1;
  const int pa = m >> 2;       // which problem
  const int r  = m & 3;        // column of A_p == row of A_p^T

  v2f_t f0, f1;
  f0.x = As[pa][k0 + 0][r]; f0.y = As[pa][k0 + 1][r];   // K-chunk 0: rows 0..3
  f1.x = As[pa][k0 + 4][r]; f1.y = As[pa][k0 + 5][r];   // K-chunk 1: rows 4..7

  v8f_t acc = {};
  acc = __builtin_amdgcn_wmma_f32_16x16x4_f32(false, f0, false, f0,
                                              (short)0, acc, false, false);
  acc = __builtin_amdgcn_wmma_f32_16x16x4_f32(false, f1, false, f1,
                                              (short)0, acc, false, false);

  // Scatter the diagonal 4x4 blocks to LDS.
  // D layout: VGPR rr -> M = rr + 8*(lane>>4), N = lane&15.
  const int n   = lane & 15;
  const int mhi = (lane >> 4) * 8;
#pragma unroll
  for (int rr = 0; rr < 8; ++rr) {
    const int M = rr + mhi;
    if ((M >> 2) == (n >> 2)) Gs[M >> 2][M & 3][n & 3] = acc[rr];
  }
  __syncthreads();

  // Per-problem 4x4 symmetric eigensolve (cyclic Jacobi), lanes 0..3.
  if (lane < 4) {
    float A4[4][4], V4[4][4];
#pragma unroll
    for (int rr = 0; rr < 4; ++rr)
#pragma unroll
      for (int cc = 0; cc < 4; ++cc) {
        A4[rr][cc] = Gs[lane][rr][cc];
        V4[rr][cc] = (rr == cc) ? 1.0f : 0.0f;
      }

    const int pp[6] = {0, 0, 0, 1, 1, 2};
    const int qq[6] = {1, 2, 3, 2, 3, 3};
    for (int sweep = 0; sweep < 10; ++sweep) {
#pragma unroll
      for (int t = 0; t < 6; ++t) {
        const int p = pp[t], q = qq[t];
        const float apq = A4[p][q];
        if (fabsf(apq) > 1e-32f) {
          const float theta = (A4[q][q] - A4[p][p]) / (2.0f * apq);
          const float sgn = (theta >= 0.0f) ? 1.0f : -1.0f;
          const float tt = sgn / (fabsf(theta) + sqrtf(theta * theta + 1.0f));
          const float cth = 1.0f / sqrtf(tt * tt + 1.0f);
          const float sth = tt * cth;
#pragma unroll
          for (int k = 0; k < 4; ++k) {
            const float akp = A4[k][p], akq = A4[k][q];
            A4[k][p] = cth * akp - sth * akq;
            A4[k][q] = sth * akp + cth * akq;
          }
#pragma unroll
          for (int k = 0; k < 4; ++k) {
            const float apk = A4[p][k], aqk = A4[q][k];
            A4[p][k] = cth * apk - sth * aqk;
            A4[q][k] = sth * apk + cth * aqk;
          }
#pragma unroll
          for (int k = 0; k < 4; ++k) {
            const float vkp = V4[k][p], vkq = V4[k][q];
            V4[k][p] = cth * vkp - sth * vkq;
            V4[k][q] = sth * vkp + cth * vkq;
          }
        }
      }
    }

    int best = 0;
    float bv = A4[0][0];
#pragma unroll
    for (int k = 1; k < 4; ++k)
      if (A4[k][k] < bv) { bv = A4[k][k]; best = k; }

    const int g = base + lane;
    const int b = g / NJ, j = g % NJ;
    const float inv = 1.0f / V4[3][best];
    float* o = out + (size_t)(b * NJ + j) * 3;
    o[0] = V4[0][best] * inv;
    o[1] = V4[1][best] * inv;
    o[2] = V4[2][best] * inv;
  }
}

extern "C" void kernel_launch(void* const* d_in, const int* in_sizes, int n_in,
                              void* d_out, int out_size, void* d_ws, size_t ws_size,
                              hipStream_t stream) {
  const float* heatmaps = (const float*)d_in[0];   // (B*C, J, 96, 96) f32
  const float* proj     = (const float*)d_in[1];   // (B, C, 3, 4) f32
  const float* conf     = (const float*)d_in[2];   // (B, C, J) f32
  const int*   Himg     = (const int*)d_in[3];     // scalar 384
  const int*   Wimg     = (const int*)d_in[4];     // scalar 384

  float* kp  = (float*)d_ws;      // (B*C, J, 2) = 8704 floats
  float* out = (float*)d_out;     // (B, J, 3)

  softargmax_kernel<<<NB * NC * NJ, T1, 0, stream>>>(heatmaps, kp, Himg, Wimg);
  dlt_wmma_kernel<<<(NB * NJ) / 4, 32, 0, stream>>>(proj, conf, kp, out);
}